// GCN_7799660609597
// MI455X (gfx1250) — compile-verified
//
#include <hip/hip_runtime.h>

typedef __attribute__((ext_vector_type(2))) float v2f;
typedef __attribute__((ext_vector_type(8))) float v8f;

// ---------------- utility kernels ----------------

__global__ void fill0_kernel(float* p, long long count) {
  long long i = (long long)blockIdx.x * blockDim.x + threadIdx.x;
  if (i < count) p[i] = 0.0f;
}

__global__ void deg_kernel(const int* __restrict__ src, const int* __restrict__ dst,
                           float* __restrict__ degO, float* __restrict__ degI, int E) {
  int e = blockIdx.x * blockDim.x + threadIdx.x;
  if (e < E) {
    atomicAdd(&degO[src[e]], 1.0f);
    atomicAdd(&degI[dst[e]], 1.0f);
  }
}

__global__ void rsq_kernel(const float* __restrict__ degO, const float* __restrict__ degI,
                           float* __restrict__ rO, float* __restrict__ rI, int n) {
  int i = blockIdx.x * blockDim.x + threadIdx.x;
  if (i < n) {
    rO[i] = rsqrtf(fmaxf(degO[i], 1.0f));
    rI[i] = rsqrtf(fmaxf(degI[i], 1.0f));
  }
}

// ---------------- GEMM1: out[n,128] = (x * rO) @ W1  (K=256) ----------------
// block = 256 threads (8 waves); block covers 128 rows; wave w covers rows w*16..w*16+15
// K chunked by 64; fp32 WMMA 16x16x4. Weight tile stored transposed [N][K] in LDS so
// each B fragment is one aligned ds_load_b64; stride 68 => conflict-free b64 access.

__global__ __launch_bounds__(256) void gemm1_kernel(
    const float* __restrict__ x, const float* __restrict__ W1,
    const float* __restrict__ rO, float* __restrict__ out, int n)
{
  __shared__ float sWt[128 * 68];  // [N=128][K=64] transposed, stride 68
  __shared__ float sA[128 * 68];   // [M=128][K=64], stride 68
  const int t = threadIdx.x;
  const int lane = t & 31;
  const int w = t >> 5;
  const int m = lane & 15;
  const int hi = lane >> 4;        // 0 or 1
  const int khalf = hi << 1;       // 0 or 2
  const int rowBase = blockIdx.x * 128;

  v8f acc[8];
#pragma unroll
  for (int i = 0; i < 8; ++i) acc[i] = 0.0f;

  for (int ko = 0; ko < 256; ko += 64) {
    // coalesced global read of W chunk, transposed store into LDS
    for (int idx = t; idx < 64 * 128; idx += 256) {
      int k = idx >> 7, c = idx & 127;
      sWt[c * 68 + k] = W1[(size_t)(ko + k) * 128 + c];
    }
    // coalesced global read of x chunk (scaled by rO)
    for (int idx = t; idx < 128 * 64; idx += 256) {
      int r = idx >> 6, c = idx & 63;
      int grow = rowBase + r;
      float v = 0.0f;
      if (grow < n) v = x[(size_t)grow * 256 + ko + c] * rO[grow];
      sA[r * 68 + c] = v;
    }
    __syncthreads();
    const float* am = &sA[(w * 16 + m) * 68];
    const float* bm = &sWt[m * 68];
#pragma unroll
    for (int kk = 0; kk < 64; kk += 4) {
      int ks = kk + khalf;
      v2f a = *reinterpret_cast<const v2f*>(am + ks);
#pragma unroll
      for (int nt = 0; nt < 8; ++nt) {
        v2f b = *reinterpret_cast<const v2f*>(bm + nt * (16 * 68) + ks);
        acc[nt] = __builtin_amdgcn_wmma_f32_16x16x4_f32(
            false, a, false, b, (short)0, acc[nt], false, false);
      }
    }
    __syncthreads();
  }

  int r0 = rowBase + w * 16 + hi * 8;
#pragma unroll
  for (int nt = 0; nt < 8; ++nt) {
#pragma unroll
    for (int j = 0; j < 8; ++j) {
      int row = r0 + j;
      if (row < n) out[(size_t)row * 128 + nt * 16 + m] = acc[nt][j];
    }
  }
}

// ---------------- GEMM2: out[n,48] = t[n,128] @ W2pad[128,48]  ----------------

__global__ __launch_bounds__(256) void gemm2_kernel(
    const float* __restrict__ tmat, const float* __restrict__ W2,
    float* __restrict__ out, int n)
{
  __shared__ float sWt[48 * 68];   // [N=48][K=64] transposed (cols 40..47 zero)
  __shared__ float sA[128 * 68];
  const int t = threadIdx.x;
  const int lane = t & 31;
  const int w = t >> 5;
  const int m = lane & 15;
  const int hi = lane >> 4;
  const int khalf = hi << 1;
  const int rowBase = blockIdx.x * 128;

  v8f acc[3];
#pragma unroll
  for (int i = 0; i < 3; ++i) acc[i] = 0.0f;

  for (int ko = 0; ko < 128; ko += 64) {
    for (int idx = t; idx < 64 * 48; idx += 256) {
      int k = idx / 48, c = idx - k * 48;
      sWt[c * 68 + k] = (c < 40) ? W2[(size_t)(ko + k) * 40 + c] : 0.0f;
    }
    for (int idx = t; idx < 128 * 64; idx += 256) {
      int r = idx >> 6, c = idx & 63;
      int grow = rowBase + r;
      sA[r * 68 + c] = (grow < n) ? tmat[(size_t)grow * 128 + ko + c] : 0.0f;
    }
    __syncthreads();
    const float* am = &sA[(w * 16 + m) * 68];
    const float* bm = &sWt[m * 68];
#pragma unroll
    for (int kk = 0; kk < 64; kk += 4) {
      int ks = kk + khalf;
      v2f a = *reinterpret_cast<const v2f*>(am + ks);
#pragma unroll
      for (int nt = 0; nt < 3; ++nt) {
        v2f b = *reinterpret_cast<const v2f*>(bm + nt * (16 * 68) + ks);
        acc[nt] = __builtin_amdgcn_wmma_f32_16x16x4_f32(
            false, a, false, b, (short)0, acc[nt], false, false);
      }
    }
    __syncthreads();
  }

  int r0 = rowBase + w * 16 + hi * 8;
#pragma unroll
  for (int nt = 0; nt < 3; ++nt) {
#pragma unroll
    for (int j = 0; j < 8; ++j) {
      int row = r0 + j;
      if (row < n) out[(size_t)row * 48 + nt * 16 + m] = acc[nt][j];
    }
  }
}

// ---------------- edge scatter (gather src row, atomic-add into dst row) ----------------
// layer 1: 128 features, 32 lanes/edge * float4 each
__global__ void scatter1_kernel(const float* __restrict__ h, const int* __restrict__ src,
                                const int* __restrict__ dst, float* __restrict__ agg,
                                long long total) {
  long long tid = (long long)blockIdx.x * blockDim.x + threadIdx.x;
  if (tid >= total) return;
  long long e = tid >> 5;
  int f4 = (int)(tid & 31) * 4;
  int s = src[e], d = dst[e];
  const float4 v = *reinterpret_cast<const float4*>(&h[(size_t)s * 128 + f4]);
  float* ap = &agg[(size_t)d * 128 + f4];
  atomicAdd(ap + 0, v.x);
  atomicAdd(ap + 1, v.y);
  atomicAdd(ap + 2, v.z);
  atomicAdd(ap + 3, v.w);
}

// layer 2: 40 valid features (stride 48 in source), 16 lanes/edge, lanes 0..9 carry float4
__global__ void scatter2_kernel(const float* __restrict__ h, const int* __restrict__ src,
                                const int* __restrict__ dst, float* __restrict__ agg,
                                long long total) {
  long long tid = (long long)blockIdx.x * blockDim.x + threadIdx.x;
  if (tid >= total) return;
  long long e = tid >> 4;
  int g = (int)(tid & 15);
  if (g >= 10) return;
  int s = src[e], d = dst[e];
  const float4 v = *reinterpret_cast<const float4*>(&h[(size_t)s * 48 + g * 4]);
  float* ap = &agg[(size_t)d * 40 + g * 4];
  atomicAdd(ap + 0, v.x);
  atomicAdd(ap + 1, v.y);
  atomicAdd(ap + 2, v.z);
  atomicAdd(ap + 3, v.w);
}

// ---------------- fused epilogue of layer 1 / prologue of layer 2 ----------------
// B[i,f] = relu(B[i,f]*rI[i] + b1[f]) * rO[i]
__global__ void act_kernel(float* __restrict__ B, const float* __restrict__ rI,
                           const float* __restrict__ rO, const float* __restrict__ b1,
                           long long total) {
  long long tid = (long long)blockIdx.x * blockDim.x + threadIdx.x;
  if (tid >= total) return;
  long long i = tid >> 7;
  int f = (int)(tid & 127);
  float v = B[tid] * rI[i] + b1[f];
  B[tid] = fmaxf(v, 0.0f) * rO[i];
}

// out[i,f] = out[i,f]*rI[i] + b2[f]
__global__ void final_kernel(float* __restrict__ out, const float* __restrict__ rI,
                             const float* __restrict__ b2, long long total) {
  long long tid = (long long)blockIdx.x * blockDim.x + threadIdx.x;
  if (tid >= total) return;
  long long i = tid / 40;
  int f = (int)(tid - i * 40);
  out[tid] = out[tid] * rI[i] + b2[f];
}

// ---------------- launcher ----------------

extern "C" void kernel_launch(void* const* d_in, const int* in_sizes, int n_in,
                              void* d_out, int out_size, void* d_ws, size_t ws_size,
                              hipStream_t stream) {
  const float* x  = (const float*)d_in[0];
  const float* W1 = (const float*)d_in[1];
  const float* b1 = (const float*)d_in[2];
  const float* W2 = (const float*)d_in[3];
  const float* b2 = (const float*)d_in[4];
  const int*   src = (const int*)d_in[5];
  const int*   dst = (const int*)d_in[6];

  const int n = in_sizes[0] / 256;
  const int E = in_sizes[5];

  float* ws = (float*)d_ws;
  float* degO = ws;                          // [n]
  float* degI = ws + (size_t)n;              // [n]
  float* rO   = ws + (size_t)2 * n;          // [n]
  float* rI   = ws + (size_t)3 * n;          // [n]
  float* A    = ws + (size_t)4 * n;          // [n*128] (h1, later h2 at stride 48)
  float* B    = A + (size_t)n * 128;         // [n*128] (agg1, later activated t)
  float* out  = (float*)d_out;               // [n*40]

  const int TB = 256;
  auto blocks = [](long long c) { return (unsigned)((c + 255) / 256); };

  // zero-init accumulators
  fill0_kernel<<<blocks((long long)2 * n), TB, 0, stream>>>(degO, (long long)2 * n);
  fill0_kernel<<<blocks((long long)n * 128), TB, 0, stream>>>(B, (long long)n * 128);
  fill0_kernel<<<blocks((long long)n * 40), TB, 0, stream>>>(out, (long long)n * 40);

  // degrees + normalization factors
  deg_kernel<<<blocks(E), TB, 0, stream>>>(src, dst, degO, degI, E);
  rsq_kernel<<<blocks(n), TB, 0, stream>>>(degO, degI, rO, rI, n);

  // layer 1
  gemm1_kernel<<<(n + 127) / 128, TB, 0, stream>>>(x, W1, rO, A, n);
  scatter1_kernel<<<blocks((long long)E * 32), TB, 0, stream>>>(A, src, dst, B,
                                                                (long long)E * 32);
  act_kernel<<<blocks((long long)n * 128), TB, 0, stream>>>(B, rI, rO, b1,
                                                            (long long)n * 128);

  // layer 2
  gemm2_kernel<<<(n + 127) / 128, TB, 0, stream>>>(B, W2, A, n);
  scatter2_kernel<<<blocks((long long)E * 16), TB, 0, stream>>>(A, src, dst, out,
                                                                (long long)E * 16);
  final_kernel<<<blocks((long long)n * 40), TB, 0, stream>>>(out, rI, b2,
                                                             (long long)n * 40);
}